// Bert4Argument_66657892433950
// MI455X (gfx1250) — compile-verified
//
#include <hip/hip_runtime.h>

typedef __attribute__((ext_vector_type(2))) float v2f;
typedef __attribute__((ext_vector_type(8))) float v8f;

#define DD     768
#define BATCH  64
#define SEQL   256
#define NLAB   200
#define NPAD   208          // 13 tiles of 16
#define KPAIRS 384          // 768/2
#define PE_ROWS 513
#define PE_TILES 33         // buffer padded to 33*16 = 528 rows

// workspace layout (float offsets)
#define OFF_SW   0
#define SZ_SW    (16384 * NPAD)
#define OFF_PEW  (OFF_SW + SZ_SW)
#define SZ_PEW   (PE_TILES * 16 * NPAD)
#define OFF_WTP0 (OFF_PEW + SZ_PEW)
#define SZ_WTP   (KPAIRS * NPAD * 2)
#define OFF_WTP1 (OFF_WTP0 + SZ_WTP)
#define OFF_BASE (OFF_WTP1 + SZ_WTP)
#define OFF_FEH  (OFF_BASE + NPAD)

// ---------------------------------------------------------------------------
// Build pair-interleaved transposed weight slabs:
//   WtP[slab][p][n] = { W[n][slab*768 + 2p], W[n][slab*768 + 2p + 1] },  n<200 else 0
// so the WMMA B operand (4x16, K pairs per lane-half) is one b64 load.
// ---------------------------------------------------------------------------
__global__ void build_wtp_kernel(const float* __restrict__ W,
                                 float* __restrict__ wtp0,
                                 float* __restrict__ wtp1) {
    int idx = blockIdx.x * blockDim.x + threadIdx.x;
    if (idx >= KPAIRS * NPAD) return;
    int p = idx / NPAD;
    int n = idx % NPAD;
    v2f a = {0.f, 0.f}, b = {0.f, 0.f};
    if (n < NLAB) {
        const float* w = W + (size_t)n * (3 * DD);
        a.x = w[2 * p];          a.y = w[2 * p + 1];
        b.x = w[DD + 2 * p];     b.y = w[DD + 2 * p + 1];
    }
    ((v2f*)wtp0)[idx] = a;
    ((v2f*)wtp1)[idx] = b;
}

// ---------------------------------------------------------------------------
// Frame-embedding slab collapses to two tiny tables (bias folded in):
//   base[c]   = b[c] + ce[0]        . W[c, 2D:3D]
//   feh[i][c] = b[c] + ce[frame[i]] . W[c, 2D:3D]
// block 0 -> base, block i>0 -> feh[i-1]
// ---------------------------------------------------------------------------
__global__ void fe_kernel(const float* __restrict__ ce,
                          const float* __restrict__ W,
                          const float* __restrict__ bias,
                          const int* __restrict__ frame,
                          float* __restrict__ base,
                          float* __restrict__ feh) {
    int row = blockIdx.x;
    int c = threadIdx.x;
    if (c >= NPAD) return;
    const float* e = (row == 0) ? ce : ce + (size_t)frame[row - 1] * DD;
    float s = 0.f;
    if (c < NLAB) {
        const float* w = W + (size_t)c * (3 * DD) + 2 * DD;
        #pragma unroll 4
        for (int d = 0; d < DD; ++d) s += e[d] * w[d];
        s += bias[c];
    }
    if (row == 0) base[c] = s;
    else          feh[(row - 1) * NPAD + c] = s;
}

// ---------------------------------------------------------------------------
// fp32 WMMA GEMM:  C[Mtiles*16 x 208] = A[M x 768] * Bt[768 x 208]
// Bt supplied pair-interleaved (KPAIRS x NPAD float2).
// Block: 13 waves (416 threads); one 16-row M tile per block; wave w owns
// N columns [16w, 16w+16). A panel (16x768, 48KB) staged in LDS.
// WMMA f32 16x16x4 lane layout:
//   A: lane = m(0..15) | half(k+2);  VGPR0/1 = K even/odd  -> ds_load_2addr_b64
//   B: lane = n(0..15) | half(k+2);  VGPR0/1 = K even/odd  -> global_load_b64
//   D: VGPR v holds row (v + 8*half), col n
// Stores are UNGUARDED: the output buffer is padded to a multiple of 16 rows
// (ragged-tail rows are written but never read), so the epilogue is a clean
// clause of 8 stores with no exec-mask juggling.
// ---------------------------------------------------------------------------
__global__ __launch_bounds__(416) void gemm_wmma_f32(
    const float* __restrict__ A,
    const float* __restrict__ BtP,   // KPAIRS*NPAD float2
    float* __restrict__ C,
    int Mvalid) {
    __shared__ float lds_a[16 * DD];

    const int tid  = threadIdx.x;
    const int row0 = blockIdx.x * 16;

    // cooperative stage of A panel (clamp ragged last tile's reads)
    for (int i = tid; i < (16 * DD) / 4; i += 416) {
        int r  = i / (DD / 4);
        int cc = i % (DD / 4);
        int gr = row0 + r;
        if (gr >= Mvalid) gr = Mvalid - 1;
        ((float4*)lds_a)[i] = *((const float4*)(A + (size_t)gr * DD) + cc);
    }
    __syncthreads();

    const int wave = tid >> 5;          // 0..12
    const int lane = tid & 31;
    const int m    = lane & 15;
    const int half = lane >> 4;
    const int n    = wave * 16 + (lane & 15);

    const float* arow = lds_a + m * DD + 2 * half;
    const v2f*   bp   = (const v2f*)BtP + half * NPAD + n;

    v8f acc = {};
    #pragma unroll 8
    for (int k0 = 0; k0 < DD; k0 += 4) {
        v2f a = *(const v2f*)(arow + k0);
        v2f b = bp[(k0 >> 1) * NPAD];
        acc = __builtin_amdgcn_wmma_f32_16x16x4_f32(
            /*neg_a=*/false, a, /*neg_b=*/false, b,
            /*c_mod=*/(short)0, acc, /*reuse_a=*/false, /*reuse_b=*/false);
    }

    float* crow = C + (size_t)(row0 + 8 * half) * NPAD + n;
    #pragma unroll
    for (int v = 0; v < 8; ++v) {
        crow[(size_t)v * NPAD] = acc[v];
    }
}

// ---------------------------------------------------------------------------
// Final assembly: out[b,l,c] = SW[b*256 + head[b,l]][c] + PEW[l-pos[b]+256][c]
//                              + (l==pos[b] ? feh[b][c] : base[c])
// c-contiguous -> fully coalesced loads and stores.
// ---------------------------------------------------------------------------
__global__ void gather_out_kernel(const float* __restrict__ SW,
                                  const float* __restrict__ PEW,
                                  const float* __restrict__ base,
                                  const float* __restrict__ feh,
                                  const int* __restrict__ head,
                                  const int* __restrict__ pos,
                                  float* __restrict__ out) {
    int idx = blockIdx.x * blockDim.x + threadIdx.x;
    if (idx >= BATCH * SEQL * NLAB) return;
    int c = idx % NLAB;
    int t = idx / NLAB;
    int l = t % SEQL;
    int b = t / SEQL;

    int h   = head[b * SEQL + l];
    int p   = pos[b];
    int rel = l - p + 256;

    float v = SW[((size_t)(b * SEQL + h)) * NPAD + c]
            + PEW[(size_t)rel * NPAD + c]
            + ((l == p) ? feh[b * NPAD + c] : base[c]);
    out[idx] = v;
}

// ---------------------------------------------------------------------------
extern "C" void kernel_launch(void* const* d_in, const int* in_sizes, int n_in,
                              void* d_out, int out_size, void* d_ws, size_t ws_size,
                              hipStream_t stream) {
    (void)in_sizes; (void)n_in; (void)out_size; (void)ws_size;

    const float* seq   = (const float*)d_in[0];   // [64,256,768]
    const float* pe    = (const float*)d_in[1];   // [513,768]
    const float* ce    = (const float*)d_in[2];   // [201,768]
    const float* W     = (const float*)d_in[3];   // [200,2304]
    const float* bias  = (const float*)d_in[4];   // [200]
    const int*   head  = (const int*)d_in[5];     // [64,256]
    const int*   frame = (const int*)d_in[6];     // [64]
    const int*   pos   = (const int*)d_in[7];     // [64]

    float* ws   = (float*)d_ws;
    float* sw   = ws + OFF_SW;
    float* pew  = ws + OFF_PEW;
    float* wtp0 = ws + OFF_WTP0;
    float* wtp1 = ws + OFF_WTP1;
    float* base = ws + OFF_BASE;
    float* feh  = ws + OFF_FEH;

    // 1) weight transpose/pack + fe tables (tiny)
    build_wtp_kernel<<<(KPAIRS * NPAD + 255) / 256, 256, 0, stream>>>(W, wtp0, wtp1);
    fe_kernel<<<BATCH + 1, 256, 0, stream>>>(ce, W, bias, frame, base, feh);

    // 2) main GEMM: SW = seq * W0^T   (16384 x 768 x 208)
    gemm_wmma_f32<<<(BATCH * SEQL) / 16, 416, 0, stream>>>(seq, wtp0, sw, BATCH * SEQL);

    // 3) position GEMM: PEW = pos_emb * W1^T   (513 x 768 x 208, padded stores)
    gemm_wmma_f32<<<PE_TILES, 416, 0, stream>>>(pe, wtp1, pew, PE_ROWS);

    // 4) gather + sum -> out [64,256,200]
    gather_out_kernel<<<(BATCH * SEQL * NLAB + 255) / 256, 256, 0, stream>>>(
        sw, pew, base, feh, head, pos, (float*)d_out);
}